// WHT3Layer_15522011808174
// MI455X (gfx1250) — compile-verified
//
#include <hip/hip_runtime.h>

// -------- CDNA5 (gfx1250) WHT via f32 WMMA + async LDS DMA --------
// H_n[i,j] = (-1)^popcount(i&j)  (Sylvester order, matches reference fwht)
// H_64  = H_4  (x) H_16   -> WMMA H16-stage + 2-stage +/- butterfly over 4 accs
// H_256 = H_16 (x) H_16   -> WMMA H16-stage + 4-stage +/- butterfly over 16 accs
//
// Each 16x16 (H16 x M) product = 4 chained V_WMMA_F32_16X16X4_F32 (K=4 each).
// H16 fragments are popcount bit-math (no loads); H16 symmetric, so the same
// per-lane values serve as A-fragment (H on left) and B-fragment (H on right).
//
// Memory strategy (23.3 TB/s HBM, 192 MB L2):
//  - x loads + t2/t3 stores are non-temporal (read-once / write-once data must
//    not evict t1 from L2);
//  - t1 stores stay RT (wr-rinse: dirty data reaches HBM as required for the
//    output, clean copy can remain in L2 for kernel 2's re-read);
//  - kernel 2 stages the contiguous 64KB (b,h)-slab through LDS; t2 is dumped
//    by GLOBAL_STORE_ASYNC_FROM_LDS_B128 overlapping phase-B WMMA work.

typedef float v2f __attribute__((ext_vector_type(2)));
typedef float v4f __attribute__((ext_vector_type(4)));
typedef float v8f __attribute__((ext_vector_type(8)));
typedef int   v4i __attribute__((vector_size(16)));   // matches builtin param type

#define V8F_ZERO {0.f, 0.f, 0.f, 0.f, 0.f, 0.f, 0.f, 0.f}

#define WMMA_F32_16x16x4(A, B, C) \
  __builtin_amdgcn_wmma_f32_16x16x4_f32(false, (A), false, (B), (short)0, (C), false, false)

#if __has_builtin(__builtin_amdgcn_global_load_async_to_lds_b128) && \
    __has_builtin(__builtin_amdgcn_global_store_async_from_lds_b128) && \
    __has_builtin(__builtin_amdgcn_s_wait_asynccnt)
#define HAVE_ASYNC_LDS 1
#else
#define HAVE_ASYNC_LDS 0
#endif

#if HAVE_ASYNC_LDS
typedef __attribute__((address_space(1))) v4i* gptr_v4i;
typedef __attribute__((address_space(3))) v4i* lptr_v4i;

__device__ __forceinline__ void async_g2l_b128(const float* g, float* l) {
  __builtin_amdgcn_global_load_async_to_lds_b128((gptr_v4i)g, (lptr_v4i)l, 0, 0);
}
__device__ __forceinline__ void async_l2g_b128(float* g, const float* l) {
  __builtin_amdgcn_global_store_async_from_lds_b128((gptr_v4i)g, (lptr_v4i)l, 0, 0);
}
#endif

__device__ __forceinline__ float hsign(int i, int j) {
  return (__builtin_popcount(i & j) & 1) ? -1.0f : 1.0f;
}

// ---------------------------------------------------------------------------
// Kernel 1: t1 = H_64 along axis H (stride 16384 floats). One wave per
// 64(h) x 16(c) tile at fixed (b, w). Pure streaming, NT input loads.
// ---------------------------------------------------------------------------
__global__ __launch_bounds__(256) void wht_h_kernel(const float* __restrict__ x,
                                                    float* __restrict__ t1) {
  const int lane = threadIdx.x & 31;
  const int wave = threadIdx.x >> 5;
  const int tidw = blockIdx.x * 8 + wave;      // 32768 waves total
  const int b    = tidw >> 10;
  const int rem  = tidw & 1023;
  const int w    = rem >> 4;
  const int cg   = rem & 15;
  const int n    = lane & 15;
  const int hi   = lane >> 4;

  const int gbase = b * 1048576 + w * 256 + cg * 16 + n;

  v2f hf[4];
#pragma unroll
  for (int s = 0; s < 4; ++s) {
    const int k = (s << 2) + (hi << 1);
    hf[s].x = hsign(n, k);
    hf[s].y = hsign(n, k + 1);
  }

  v8f res[4];
#pragma unroll
  for (int e = 0; e < 4; ++e) {
    v8f acc = V8F_ZERO;
#pragma unroll
    for (int s = 0; s < 4; ++s) {
      const int row = e * 16 + (s << 2) + (hi << 1);
      v2f bf;
      bf.x = __builtin_nontemporal_load(&x[gbase + row * 16384]);
      bf.y = __builtin_nontemporal_load(&x[gbase + (row + 1) * 16384]);
      acc = WMMA_F32_16x16x4(hf[s], bf, acc);
    }
    res[e] = acc;
  }
  {
    const v8f u0 = res[0] + res[1], u1 = res[0] - res[1];
    const v8f u2 = res[2] + res[3], u3 = res[2] - res[3];
    res[0] = u0 + u2; res[1] = u1 + u3; res[2] = u0 - u2; res[3] = u1 - u3;
  }
#pragma unroll
  for (int a = 0; a < 4; ++a)
#pragma unroll
    for (int j = 0; j < 8; ++j) {
      const int h = a * 16 + j + (hi << 3);
      t1[gbase + h * 16384] = res[a][j];       // RT store: rinse to HBM, stay in L2
    }
}

// ---------------------------------------------------------------------------
// Kernel 2 (fused): t2 = H_64 along W, t3 = H_256 along C, one (b,h) slab
// (16384 contiguous floats) staged in a 64 KiB swizzled LDS tile.
// XOR swizzle f: w[3:0] -> bank bits [5:2]; all access phases conflict-free.
// ---------------------------------------------------------------------------
__device__ __forceinline__ int swz(int w, int c) {
  const int f = ((w & 1) << 2)   // w0 -> bit2
              | ((w & 4) << 1)   // w2 -> bit3
              | ((w & 2) << 3)   // w1 -> bit4 (phase-A read halves: w vs w+2)
              | ((w & 8) << 2);  // w3 -> bit5 (phase-A/B write halves: w vs w+8)
  return (w << 8) + (c ^ f);
}

__global__ __launch_bounds__(128) void wht_wc_kernel(const float* __restrict__ t1,
                                                     float* __restrict__ t2,
                                                     float* __restrict__ t3) {
  __shared__ float tile[64 * 256];             // exactly 64 KiB

  const int tid  = threadIdx.x;
  const int lane = tid & 31;
  const int wv   = tid >> 5;                   // 4 waves / block
  const int n    = lane & 15;
  const int hi   = lane >> 4;
  const int base = blockIdx.x * 16384;         // blockIdx = b*64 + h

  // ---- fill: contiguous slab -> swizzled LDS tile
#if HAVE_ASYNC_LDS
  {
    const float* src = t1 + base;
#pragma unroll
    for (int i = 0; i < 32; ++i) {
      const int p   = tid + i * 128;           // float4 index; rows never straddled
      const int row = p >> 6;
      const int col = (p & 63) << 2;
      async_g2l_b128(src + (p << 2), &tile[swz(row, col)]);
    }
    __builtin_amdgcn_s_wait_asynccnt(0);
  }
#else
  {
    const float* src = t1 + base;
#pragma unroll
    for (int i = 0; i < 32; ++i) {
      const int p   = tid + i * 128;
      const v4f  v  = *(const v4f*)(src + (p << 2));
      const int row = p >> 6;
      const int col = (p & 63) << 2;
      *(v4f*)(&tile[swz(row, col)]) = v;
    }
  }
#endif
  __syncthreads();

  v2f hf[4];
#pragma unroll
  for (int s = 0; s < 4; ++s) {
    const int k = (s << 2) + (hi << 1);
    hf[s].x = hsign(n, k);
    hf[s].y = hsign(n, k + 1);
  }

  // ---- Phase A: H_64 along W. Wave wv owns channel groups {wv, wv+4, wv+8, wv+12}
  //      (disjoint columns per wave). Results go back into the tile only.
#pragma unroll 1
  for (int t = 0; t < 4; ++t) {
    const int cg = wv + (t << 2);
    const int cb = cg * 16 + n;
    v8f res[4];
#pragma unroll
    for (int e = 0; e < 4; ++e) {
      v8f acc = V8F_ZERO;
#pragma unroll
      for (int s = 0; s < 4; ++s) {
        const int wr = e * 16 + (s << 2) + (hi << 1);
        v2f bf;
        bf.x = tile[swz(wr, cb)];
        bf.y = tile[swz(wr + 1, cb)];
        acc = WMMA_F32_16x16x4(hf[s], bf, acc);
      }
      res[e] = acc;
    }
    {
      const v8f u0 = res[0] + res[1], u1 = res[0] - res[1];
      const v8f u2 = res[2] + res[3], u3 = res[2] - res[3];
      res[0] = u0 + u2; res[1] = u1 + u3; res[2] = u0 - u2; res[3] = u1 - u3;
    }
#pragma unroll
    for (int a = 0; a < 4; ++a)
#pragma unroll
      for (int j = 0; j < 8; ++j) {
        const int wq = a * 16 + j + (hi << 3);
        tile[swz(wq, cb)] = res[a][j];
      }
  }
  __syncthreads();                             // tile now holds t2

  // ---- t2 dump: wave wv streams its own rows w0..w0+15 (coalesced 512B).
  //      Async path overlaps the DMA with phase-B WMMA work below.
  const int w0 = wv << 4;
#if HAVE_ASYNC_LDS
  {
    float* dst2 = t2 + base;
#pragma unroll
    for (int i = 0; i < 32; ++i) {
      const int p4  = (wv << 10) + (i << 5) + lane;    // float4 index
      const int row = p4 >> 6;
      const int col = (p4 & 63) << 2;
      async_l2g_b128(dst2 + (p4 << 2), &tile[swz(row, col)]);
    }
  }
#else
  {
    float* dst2 = t2 + base;
#pragma unroll
    for (int i = 0; i < 32; ++i) {
      const int p4  = (wv << 10) + (i << 5) + lane;
      const int row = p4 >> 6;
      const int col = (p4 & 63) << 2;
      const v4f v = *(const v4f*)(&tile[swz(row, col)]);
      __builtin_nontemporal_store(v, (v4f*)(dst2 + (p4 << 2)));
    }
  }
#endif

  // ---- Phase B: H_256 along C over rows w0..w0+15 (data in A-frag, H16 in
  //      B-frag). Reads only this wave's rows -> safe vs. the async reads above.
  v8f acc[16];
#pragma unroll
  for (int e = 0; e < 16; ++e) {
    v8f z = V8F_ZERO;
    acc[e] = z;
  }
#pragma unroll
  for (int e = 0; e < 16; ++e) {
#pragma unroll
    for (int s = 0; s < 4; ++s) {
      const int k  = (s << 2) + (hi << 1);
      const v2f af = *(const v2f*)(&tile[swz(w0 + n, e * 16 + k)]);  // 8B aligned
      acc[e] = WMMA_F32_16x16x4(af, hf[s], acc[e]);
    }
  }
#pragma unroll
  for (int bit = 1; bit < 16; bit <<= 1) {
#pragma unroll
    for (int e = 0; e < 16; ++e) {
      if ((e & bit) == 0) {
        const v8f xx = acc[e];
        const v8f yy = acc[e | bit];
        acc[e]       = xx + yy;
        acc[e | bit] = xx - yy;
      }
    }
  }

#if HAVE_ASYNC_LDS
  __builtin_amdgcn_s_wait_asynccnt(0);         // t2 DMA done before overwriting rows
#endif

  // ---- t3: write back into own tile rows (exclusive per wave; DS ops are
  //      in-order within a wave), then coalesced NT dump.
#pragma unroll
  for (int a = 0; a < 16; ++a)
#pragma unroll
    for (int j = 0; j < 8; ++j) {
      const int wq = w0 + j + (hi << 3);
      tile[swz(wq, a * 16 + n)] = acc[a][j];
    }
  {
    float* dst3 = t3 + base;
#pragma unroll
    for (int i = 0; i < 32; ++i) {
      const int p4  = (wv << 10) + (i << 5) + lane;
      const int row = p4 >> 6;
      const int col = (p4 & 63) << 2;
      const v4f v = *(const v4f*)(&tile[swz(row, col)]);
      __builtin_nontemporal_store(v, (v4f*)(dst3 + (p4 << 2)));
    }
  }
}

// ---------------------------------------------------------------------------
extern "C" void kernel_launch(void* const* d_in, const int* in_sizes, int n_in,
                              void* d_out, int out_size, void* d_ws, size_t ws_size,
                              hipStream_t stream) {
  (void)in_sizes; (void)n_in; (void)out_size; (void)d_ws; (void)ws_size;

  const float* x   = (const float*)d_in[0];
  float*       out = (float*)d_out;
  const long long N = 32LL * 64 * 64 * 256;    // 33,554,432 elements per tensor
  float* t1 = out;
  float* t2 = out + N;
  float* t3 = out + 2 * N;

  wht_h_kernel<<<4096, 256, 0, stream>>>(x, t1);
  wht_wc_kernel<<<2048, 128, 0, stream>>>(t1, t2, t3);
}